// AssignBoxes_36807869727184
// MI455X (gfx1250) — compile-verified
//
#include <hip/hip_runtime.h>
#include <hip/hip_bf16.h>

#define NUM_CLASSES 80
#define NB 8
#define NG 64
#define NP 32768
#define IOU_EPS 1e-5f

typedef __attribute__((ext_vector_type(2))) float v2f;
typedef __attribute__((ext_vector_type(8))) float v8f;

// ---------------------------------------------------------------------------
// Kernel 1: per (b,g) argmax over P of masked IoU (first-occurrence tie-break,
// matching jnp.argmax). One 256-thread block per (b,g); pr slice stays in L2.
// ---------------------------------------------------------------------------
__global__ void AssignBoxes_argmax_kernel(const float* __restrict__ gt,
                                          const float* __restrict__ pr,
                                          int* __restrict__ bestp) {
    __shared__ float sval[256];
    __shared__ int   sidx[256];
    const int bg = blockIdx.x;              // b*NG + g
    const float* grow = gt + (size_t)bg * 6;
    const float cx = grow[0], cy = grow[1], gw = grow[2], gh = grow[3];
    const float valid = (cx != -1.0f) ? 1.0f : 0.0f;
    const float ga  = gw * gh;
    const float gy1 = cy - gh * 0.5f, gx1 = cx - gw * 0.5f;
    const float gy2 = cy + gh * 0.5f, gx2 = cx + gw * 0.5f;

    float bv = -1.0f;
    int   bi = 0;
    for (int p = threadIdx.x; p < NP; p += 256) {
        const float4 pb = *(const float4*)(pr + (size_t)p * 4);
        const float pa  = pb.z * pb.w;
        const float py1 = pb.y - pb.w * 0.5f, px1 = pb.x - pb.z * 0.5f;
        const float py2 = pb.y + pb.w * 0.5f, px2 = pb.x + pb.z * 0.5f;
        const float ih = fmaxf(0.0f, fminf(gy2, py2) - fmaxf(gy1, py1));
        const float iw = fmaxf(0.0f, fminf(gx2, px2) - fmaxf(gx1, px1));
        const float inter = iw * ih;
        const float iou = inter / (ga + pa - inter + IOU_EPS) * valid;
        if (iou > bv) { bv = iou; bi = p; }   // strictly > keeps earliest p
    }
    sval[threadIdx.x] = bv;
    sidx[threadIdx.x] = bi;
    __syncthreads();
    for (int s = 128; s > 0; s >>= 1) {
        if (threadIdx.x < (unsigned)s) {
            const float ov = sval[threadIdx.x + s];
            const int   oi = sidx[threadIdx.x + s];
            const float mv = sval[threadIdx.x];
            const int   mi = sidx[threadIdx.x];
            if (ov > mv || (ov == mv && oi < mi)) {
                sval[threadIdx.x] = ov;
                sidx[threadIdx.x] = oi;
            }
        }
        __syncthreads();
    }
    if (threadIdx.x == 0) bestp[bg] = sidx[0];
}

// ---------------------------------------------------------------------------
// Kernel 2: fused assignment. 256 threads = 8 waves; each wave owns a
// 16-prior tile and runs the ('bgp,bg->bp') einsum as 16 fully-unrolled
// chained V_WMMA_F32_16X16X4_F32 ops (A = padded stats rows from LDS,
// B = w flags built on the fly from IoU). Lane l serves A-row (l&15) and
// B-column (l&15); the two lane halves split K. D rows 0..4 in lanes 0..15
// are W, S_cx, S_cy, S_lw, S_lh. One-hot output staged via LDS, written as
// coalesced b128 stores.
// ---------------------------------------------------------------------------
__global__ void AssignBoxes_assign_kernel(const float* __restrict__ gt,
                                          const float* __restrict__ pr,
                                          const int* __restrict__ bestp,
                                          float* __restrict__ out_cls,
                                          float* __restrict__ out_loc,
                                          float* __restrict__ out_mask) {
    __shared__ float s_cx[NG], s_cy[NG], s_lw[NG], s_lh[NG];
    __shared__ float s_x1[NG], s_y1[NG], s_x2[NG], s_y2[NG];
    __shared__ float s_area[NG], s_valid[NG], s_label[NG];
    __shared__ int   s_best[NG];
    __shared__ float s_statsA[16 * 65];     // stride 65 keeps banks spread
    __shared__ int   s_cls[128];

    const int tid    = threadIdx.x;
    const int b      = blockIdx.y;
    const int p_base = blockIdx.x * 128;

    // CDNA5 prefetch of this block's pr tile (global_prefetch_b8)
    __builtin_prefetch(pr + (size_t)(p_base + tid) * 4, 0, 1);

    if (tid < NG) {
        const float* row = gt + ((size_t)b * NG + tid) * 6;
        const float cx = row[0], cy = row[1], w = row[2], h = row[3];
        const float lab = row[4], conf = row[5];
        s_cx[tid] = cx; s_cy[tid] = cy;
        s_lw[tid] = (w > 0.0f) ? logf(fmaxf(w, 1e-20f)) : 0.0f;
        s_lh[tid] = (h > 0.0f) ? logf(fmaxf(h, 1e-20f)) : 0.0f;
        s_x1[tid] = cx - w * 0.5f; s_y1[tid] = cy - h * 0.5f;
        s_x2[tid] = cx + w * 0.5f; s_y2[tid] = cy + h * 0.5f;
        s_area[tid]  = w * h;
        s_valid[tid] = (cx != -1.0f) ? 1.0f : 0.0f;
        s_label[tid] = lab;
        s_best[tid]  = (conf > 0.0f) ? bestp[b * NG + tid] : -1;
    }
    __syncthreads();
    // A-matrix table: rows {1, cx, cy, log w, log h, 0...0} x K=64
    for (int idx = tid; idx < 16 * NG; idx += 256) {
        const int m = idx >> 6, g = idx & 63;
        float v = 0.0f;
        if      (m == 0) v = 1.0f;
        else if (m == 1) v = s_cx[g];
        else if (m == 2) v = s_cy[g];
        else if (m == 3) v = s_lw[g];
        else if (m == 4) v = s_lh[g];
        s_statsA[m * 65 + g] = v;
    }
    __syncthreads();

    const int lane = tid & 31;
    const int wv   = tid >> 5;
    const int hh   = lane >> 4;             // lane half selects K pair (2*hh, 2*hh+1) of each 4
    const int j    = lane & 15;             // stat row (A) == prior col (B)
    const int p    = p_base + wv * 16 + j;

    const float4 pb = *(const float4*)(pr + (size_t)p * 4);
    const float pcx = pb.x, pcy = pb.y, pw = pb.z, ph = pb.w;
    const float pa  = pw * ph;
    const float py1 = pcy - ph * 0.5f, px1 = pcx - pw * 0.5f;
    const float py2 = pcy + ph * 0.5f, px2 = pcx + pw * 0.5f;

    v8f acc = {0.f, 0.f, 0.f, 0.f, 0.f, 0.f, 0.f, 0.f};
    int gth = -1, gbe = -1, ign = 0;

#pragma unroll
    for (int k = 0; k < 16; ++k) {
        const int g0 = 4 * k + 2 * hh;
        float wgt[2];
#pragma unroll
        for (int s = 0; s < 2; ++s) {
            const int g = g0 + s;
            const float ihh = fmaxf(0.0f, fminf(s_y2[g], py2) - fmaxf(s_y1[g], py1));
            const float iww = fmaxf(0.0f, fminf(s_x2[g], px2) - fmaxf(s_x1[g], px1));
            const float inter = iww * ihh;
            const float iou = inter / (s_area[g] + pa - inter + IOU_EPS) * s_valid[g];
            const int th = (iou >= 0.5f);
            const int be = (s_best[g] == p);
            if (th) gth = max(gth, g);
            if (be) gbe = max(gbe, g);
            ign |= (iou >= 0.4f) & (iou < 0.5f);
            wgt[s] = (float)(th + be);
        }
        v2f A, Bm;
        A.x  = s_statsA[j * 65 + g0];
        A.y  = s_statsA[j * 65 + g0 + 1];
        Bm.x = wgt[0];
        Bm.y = wgt[1];
        acc = __builtin_amdgcn_wmma_f32_16x16x4_f32(false, A, false, Bm,
                                                    (short)0, acc, false, false);
    }

    // merge per-prior flags across the two lane halves (same p in l and l^16)
    gth  = max(gth, __shfl_xor(gth, 16, 32));
    gbe  = max(gbe, __shfl_xor(gbe, 16, 32));
    ign |= __shfl_xor(ign, 16, 32);

    if (hh == 0) {
        const float W   = acc[0];
        const float Scx = acc[1];
        const float Scy = acc[2];
        const float Slw = acc[3];
        const float Slh = acc[4];
        const int win = (gbe >= 0) ? gbe : gth;
        int   cls;
        float mask;
        if (win >= 0) { cls = (int)s_label[win]; mask = 0.0f; }
        else          { cls = NUM_CLASSES;       mask = 1.0f; }
        if (ign) mask = -1.0f;

        const float hcx = (Scx - W * pcx) / pw;
        const float hcy = (Scy - W * pcy) / ph;
        const float hw  = Slw - W * logf(pw);
        const float hhv = Slh - W * logf(ph);

        const size_t bp = (size_t)b * NP + p;
        *(float4*)(out_loc + bp * 4) = make_float4(hcx, hcy, hw, hhv);
        out_mask[bp] = mask;
        s_cls[wv * 16 + j] = cls;
    }
    __syncthreads();

    // coalesced one-hot: 128 priors * 80 classes as b128 stores
    // (base is a multiple of 80 floats = 320B, each vec4 16B aligned)
    float4* cls4 = (float4*)(out_cls + ((size_t)b * NP + p_base) * NUM_CLASSES);
    for (int idx4 = tid; idx4 < 128 * (NUM_CLASSES / 4); idx4 += 256) {
        const int pl = idx4 / (NUM_CLASSES / 4);
        const int c0 = (idx4 - pl * (NUM_CLASSES / 4)) * 4;
        const int cid = s_cls[pl];
        float4 v;
        v.x = (cid == c0    ) ? 1.0f : 0.0f;
        v.y = (cid == c0 + 1) ? 1.0f : 0.0f;
        v.z = (cid == c0 + 2) ? 1.0f : 0.0f;
        v.w = (cid == c0 + 3) ? 1.0f : 0.0f;
        cls4[idx4] = v;
    }
}

extern "C" void kernel_launch(void* const* d_in, const int* in_sizes, int n_in,
                              void* d_out, int out_size, void* d_ws, size_t ws_size,
                              hipStream_t stream) {
    const float* gt = (const float*)d_in[0];   // (B,G,6) f32
    const float* pr = (const float*)d_in[1];   // (B,P,4) f32; reference uses pr[0]
    float* out      = (float*)d_out;
    float* out_cls  = out;                                            // (B,P,80)
    float* out_loc  = out_cls + (size_t)NB * NP * NUM_CLASSES;        // (B,P,4)
    float* out_mask = out_loc + (size_t)NB * NP * 4;                  // (B,P,1)
    int* bestp = (int*)d_ws;                                          // B*G ints

    AssignBoxes_argmax_kernel<<<NB * NG, 256, 0, stream>>>(gt, pr, bestp);

    dim3 grid(NP / 128, NB);
    AssignBoxes_assign_kernel<<<grid, 256, 0, stream>>>(gt, pr, bestp,
                                                        out_cls, out_loc, out_mask);
}